// ClusterSeedClsWithFilter_32555852103944
// MI455X (gfx1250) — compile-verified
//
#include <hip/hip_runtime.h>
#include <stdint.h>

// Problem geometry
#define HH 1024
#define WW 2048
#define NPIX (HH * WW)          // 2097152
#define NCH 7

// Reference constants
#define MAX_COUNT      200
#define MIN_PIXEL      160
#define MIN_INST_PIXEL 160
#define THRESHOLD      0.5f
#define MIN_SEED_THRESH 0.5f
#define INST_RATIO     0.5f
#define DIST_THRESH    0.5f

// Persistent-kernel geometry
#define NB 64                   // persistent blocks (co-resident; one grid barrier domain)
#define NT 1024                 // threads per block (32 wave32 waves)
#define CHUNK (NPIX / NB)       // 32768 contiguous pixels per block stripe
#define VSTEP (NT * 4)          // pixels consumed per block per vector step

struct Ctrl {
  int arrive;         // grid barrier arrive counter
  int gen;            // grid barrier generation
  int done;
  int iter_ran;       // a proposal/apply phase has run and awaits finalization
  int count;          // next cluster id (starts at 1, as in reference)
  int uncl_count;     // sum(unclustered)
  int seed_idx;
  float cx, cy, sx, sy;
  int prop_sum;
  int uncl_in_prop;
  int uncl_next;
  int prev[MAX_COUNT];
  int now[MAX_COUNT];
  int bad[MAX_COUNT];
  float pmax[NB];
  int   pidx[NB];
};

union FL4 { float4 v; float   a[4]; };
union UC4 { uchar4 v; uint8_t a[4]; unsigned w; };

// ---------------- grid barrier (64 blocks, software, device-scope atomics) --
__device__ __forceinline__ void gbar(Ctrl* c) {
  __syncthreads();
  if (threadIdx.x == 0) {
    __threadfence();  // release my block's global writes to L2
    int g = __atomic_load_n(&c->gen, __ATOMIC_ACQUIRE);
    int a = __atomic_add_fetch(&c->arrive, 1, __ATOMIC_ACQ_REL);
    if (a == NB) {
      __atomic_store_n(&c->arrive, 0, __ATOMIC_RELAXED);
      __atomic_fetch_add(&c->gen, 1, __ATOMIC_RELEASE);
    } else {
      while (__atomic_load_n(&c->gen, __ATOMIC_ACQUIRE) == g) {
        __builtin_amdgcn_s_sleep(1);
      }
    }
  }
  __syncthreads();
}

// ---------------- block-wide integer sum (wave32 shuffles) ------------------
__device__ __forceinline__ int blockReduceSum(int v, int* s) {
  int lane = threadIdx.x & 31, wid = threadIdx.x >> 5;
#pragma unroll
  for (int o = 16; o > 0; o >>= 1) v += __shfl_down(v, o, 32);
  if (lane == 0) s[wid] = v;
  __syncthreads();
  v = (threadIdx.x < 32) ? s[threadIdx.x] : 0;   // exactly 32 waves (NT=1024)
  if (wid == 0) {
#pragma unroll
    for (int o = 16; o > 0; o >>= 1) v += __shfl_down(v, o, 32);
  }
  __syncthreads();
  return v;  // valid in thread 0
}

// ------------- block argmax (value, first-index tie-break) -> ctrl partials
__device__ __forceinline__ void argmaxToCtrl(float bv, int bi, float* s_f, int* s_ii,
                                             Ctrl* c, int bid) {
#pragma unroll
  for (int o = 16; o > 0; o >>= 1) {
    float ov = __shfl_down(bv, o, 32);
    int   oi = __shfl_down(bi, o, 32);
    if (ov > bv || (ov == bv && oi < bi)) { bv = ov; bi = oi; }
  }
  int lane = threadIdx.x & 31, wid = threadIdx.x >> 5;
  if (lane == 0) { s_f[wid] = bv; s_ii[wid] = bi; }
  __syncthreads();
  if (wid == 0) {
    bv = s_f[lane]; bi = s_ii[lane];
#pragma unroll
    for (int o = 16; o > 0; o >>= 1) {
      float ov = __shfl_down(bv, o, 32);
      int   oi = __shfl_down(bi, o, 32);
      if (ov > bv || (ov == bv && oi < bi)) { bv = ov; bi = oi; }
    }
    if (lane == 0) { c->pmax[bid] = bv; c->pidx[bid] = bi; }
  }
  __syncthreads();
}

// ---------------- kernel 0: zero control state ------------------------------
__global__ void k_init(Ctrl* c) {
  int t = threadIdx.x;
  if (t == 0) {
    c->arrive = 0; c->gen = 0; c->done = 0; c->iter_ran = 0;
    c->count = 1; c->uncl_count = 0; c->seed_idx = 0;
    c->prop_sum = 0; c->uncl_in_prop = 0; c->uncl_next = 0;
  }
  for (int j = t; j < MAX_COUNT; j += (int)blockDim.x) {
    c->prev[j] = 0; c->now[j] = 0; c->bad[j] = 0;
  }
}

// ---------------- kernel 1: per-pixel preprocess (HBM-bound, full grid) -----
__global__ __launch_bounds__(256) void k_pre(const float* __restrict__ pred, Ctrl* c,
                                             float* __restrict__ sex, float* __restrict__ sey,
                                             float* __restrict__ sd,
                                             uint8_t* __restrict__ uncl,
                                             uint8_t* __restrict__ inst) {
  int i = blockIdx.x * 256 + threadIdx.x;
  int x = i & (WW - 1);
  int y = i >> 11;  // WW == 2048
  float p0 = pred[i];
  float p1 = pred[NPIX + i];
  float p5 = pred[5 * NPIX + i];
  float p6 = pred[6 * NPIX + i];
  float xm = (float)x * (2.0f / 2047.0f);   // linspace(0,2,2048)
  float ym = (float)y * (1.0f / 1023.0f);   // linspace(0,1,1024)
  sex[i] = tanhf(p0) + xm;
  sey[i] = tanhf(p1) + ym;
  float s = 1.0f / (1.0f + expf(p5 - p6));  // softmax([p5,p6])[1]
  sd[i] = s;
  int mm = (s > MIN_SEED_THRESH) ? 1 : 0;
  uncl[i] = (uint8_t)mm;
  inst[i] = 0;
  unsigned long long b = __ballot(mm);      // wave32: low 32 bits
  if ((threadIdx.x & 31) == 0) atomicAdd(&c->uncl_count, (int)__popcll(b));
}

// ---------------- kernel 2: persistent clustering loop + postprocess --------
__global__ __launch_bounds__(NT) void k_cluster(const float* __restrict__ pred,
                                                float* __restrict__ out,
                                                Ctrl* __restrict__ c,
                                                const float* __restrict__ sex,
                                                const float* __restrict__ sey,
                                                const float* __restrict__ sd,
                                                uint8_t* __restrict__ uncl,
                                                uint8_t* __restrict__ prop,
                                                uint8_t* __restrict__ inst) {
  const int tid = threadIdx.x;
  const int bid = blockIdx.x;
  const int base = bid * CHUNK;

  __shared__ int   s_i[32];
  __shared__ float s_f[32];
  __shared__ int   s_ii[32];
  __shared__ float l_pmax[NB];
  __shared__ int   l_pidx[NB];
  __shared__ int   s_hist[MAX_COUNT];

  // ---- Initial argmax partials over scores = seed * unclustered ------------
  {
    float bv = -1.0f; int bi = 0;
    for (int k = 0; k < CHUNK; k += VSTEP) {
      int i = base + k + (tid << 2);
      FL4 es; es.v = *(const float4*)(sd + i);
      UC4 u;  u.v  = *(const uchar4*)(uncl + i);
#pragma unroll
      for (int j = 0; j < 4; ++j) {
        float v = u.a[j] ? es.a[j] : 0.0f;
        if (v > bv) { bv = v; bi = i + j; }   // increasing index => first max
      }
    }
    argmaxToCtrl(bv, bi, s_f, s_ii, c, bid);
  }
  gbar(c);

  for (;;) {
    // ---- Phase B: block 0 finalizes previous iteration, evaluates loop cond,
    //      reduces the argmax partials (staged via gfx1250 async-to-LDS).
    if (bid == 0) {
#if defined(__AMDGCN__)
      if (tid < NB) {
        uint32_t lf = (uint32_t)(uintptr_t)&l_pmax[tid];
        uint32_t li = (uint32_t)(uintptr_t)&l_pidx[tid];
        const float* gp = &c->pmax[tid];
        const int*   gi = &c->pidx[tid];
        asm volatile(
            "global_load_async_to_lds_b32 %0, %1, off scope:SCOPE_DEV\n\t"
            "global_load_async_to_lds_b32 %2, %3, off scope:SCOPE_DEV"
            :: "v"(lf), "v"(gp), "v"(li), "v"(gi)
            : "memory");
      }
      asm volatile("s_wait_asynccnt 0x0" ::: "memory");
#else
      if (tid < NB) { l_pmax[tid] = c->pmax[tid]; l_pidx[tid] = c->pidx[tid]; }
#endif
      __syncthreads();
      if (tid == 0) {
        if (c->iter_ran) {  // finalize the iteration whose C/E phases just ran
          int ps = c->prop_sum, up = c->uncl_in_prop;
          int denom = ps > 1 ? ps : 1;
          int acc = (ps > MIN_INST_PIXEL) &&
                    ((float)up / (float)denom > INST_RATIO);
          if (acc) { c->prev[c->count] = ps; c->count = c->count + 1; }
          c->uncl_count = c->uncl_next;
        }
        c->prop_sum = 0; c->uncl_in_prop = 0; c->uncl_next = 0;
        int done = (c->uncl_count <= MIN_PIXEL) || (c->count >= MAX_COUNT);
        if (!done) {
          float bv2 = -1.0f; int bi2 = 0;
          for (int j = 0; j < NB; ++j) {
            float v = l_pmax[j]; int ii = l_pidx[j];
            if (v > bv2 || (v == bv2 && ii < bi2)) { bv2 = v; bi2 = ii; }
          }
          if (bv2 < THRESHOLD) {
            done = 1;                       // stop_now: discard, keep state
          } else {
            c->seed_idx = bi2;
            c->cx = sex[bi2];
            c->cy = sey[bi2];
            c->sx = expf(pred[2 * NPIX + bi2] * 10.0f);
            c->sy = expf(pred[3 * NPIX + bi2] * 10.0f);
            c->iter_ran = 1;
          }
        }
        c->done = done;
      }
    }
    gbar(c);
    if (c->done) break;

    // ---- Phase C (fused): proposal + uncl update + integer sums +
    //      NEXT iteration's argmax partials, single vectorized pass ---------
    {
      const int   sidx = c->seed_idx;
      const float cx = c->cx, cy = c->cy, sxv = c->sx, syv = c->sy;
      int ps = 0, up = 0, rem = 0;
      float bv = -1.0f; int bi = 0;
      for (int k = 0; k < CHUNK; k += VSTEP) {
        int i = base + k + (tid << 2);
        if (k + VSTEP < CHUNK) {
          __builtin_prefetch(sex + i + VSTEP, 0, 0);   // global_prefetch_b8
          __builtin_prefetch(sey + i + VSTEP, 0, 0);
        }
        FL4 ex; ex.v = *(const float4*)(sex + i);
        FL4 ey; ey.v = *(const float4*)(sey + i);
        FL4 es; es.v = *(const float4*)(sd + i);
        UC4 u;  u.v  = *(const uchar4*)(uncl + i);
        UC4 p4, n4;
#pragma unroll
        for (int j = 0; j < 4; ++j) {
          int   idx = i + j;
          float dx = ex.a[j] - cx;
          float dy = ey.a[j] - cy;
          float d2 = dx * dx * sxv + dy * dy * syv;
          int p  = (expf(-d2) > DIST_THRESH) && (es.a[j] > MIN_SEED_THRESH);
          int uu = u.a[j];
          int nu = (p || idx == sidx) ? 0 : uu;
          ps  += p;
          up  += (p && uu && (idx != sidx)) ? 1 : 0;
          rem += nu;
          float score = nu ? es.a[j] : 0.0f;
          if (score > bv) { bv = score; bi = idx; }
          p4.a[j] = (uint8_t)p;
          n4.a[j] = (uint8_t)nu;
        }
        *(uchar4*)(prop + i) = p4.v;
        *(uchar4*)(uncl + i) = n4.v;
      }
      ps  = blockReduceSum(ps, s_i);
      if (tid == 0 && ps)  atomicAdd(&c->prop_sum, ps);
      up  = blockReduceSum(up, s_i);
      if (tid == 0 && up)  atomicAdd(&c->uncl_in_prop, up);
      rem = blockReduceSum(rem, s_i);
      if (tid == 0 && rem) atomicAdd(&c->uncl_next, rem);
      argmaxToCtrl(bv, bi, s_f, s_ii, c, bid);
    }
    gbar(c);

    // ---- Phase E: inst assignment, only on accepted iterations -------------
    {
      int psum = c->prop_sum, upv = c->uncl_in_prop;
      int acc = (psum > MIN_INST_PIXEL) &&
                ((float)upv / (float)(psum > 1 ? psum : 1) > INST_RATIO);
      if (acc) {
        int cur = c->count;   // stable until next Phase B
        for (int k = 0; k < CHUNK; k += VSTEP) {
          int i = base + k + (tid << 2);
          UC4 p; p.v = *(const uchar4*)(prop + i);
          if (p.w) {          // skip untouched 4-pixel words (proposals local)
            UC4 t; t.v = *(const uchar4*)(inst + i);
#pragma unroll
            for (int j = 0; j < 4; ++j)
              if (p.a[j]) t.a[j] = (uint8_t)cur;
            *(uchar4*)(inst + i) = t.v;
          }
        }
      }
    }
    gbar(c);
  }

  // ---- Postprocess: bincount -> bad[] -> filtered float output -------------
  for (int j = tid; j < MAX_COUNT; j += NT) s_hist[j] = 0;
  __syncthreads();
  int z = 0;
  for (int k = 0; k < CHUNK; k += VSTEP) {
    int i = base + k + (tid << 2);
    UC4 t; t.v = *(const uchar4*)(inst + i);
    if (t.w) {
#pragma unroll
      for (int j = 0; j < 4; ++j) {
        int v = t.a[j];
        if (v) atomicAdd(&s_hist[v], 1);
        else   z++;
      }
    } else {
      z += 4;                        // avoid LDS-atomic contention on bin 0
    }
  }
  __syncthreads();
  z = blockReduceSum(z, s_i);
  if (tid == 0 && z) atomicAdd(&c->now[0], z);
  for (int j = tid; j < MAX_COUNT; j += NT) {
    if (j > 0 && s_hist[j]) atomicAdd(&c->now[j], s_hist[j]);
  }
  gbar(c);

  if (bid == 0 && tid < MAX_COUNT) {
    int nowc = c->now[tid], prevc = c->prev[tid];
    float ratio = (float)nowc / (float)(prevc > 1 ? prevc : 1);
    int b = (nowc != prevc) && (nowc > 0) &&
            ((nowc < MIN_INST_PIXEL * 3) || (ratio < INST_RATIO));
    if (tid == 0) b = 0;
    c->bad[tid] = b;
  }
  gbar(c);

  for (int j = tid; j < MAX_COUNT; j += NT) s_hist[j] = c->bad[j];
  __syncthreads();
  for (int k = 0; k < CHUNK; k += VSTEP) {
    int i = base + k + (tid << 2);
    UC4 t; t.v = *(const uchar4*)(inst + i);
    FL4 o;
#pragma unroll
    for (int j = 0; j < 4; ++j) {
      int v = t.a[j];
      o.a[j] = s_hist[v] ? 0.0f : (float)v;
    }
    *(float4*)(out + i) = o.v;
  }
}

// ---------------------------------------------------------------------------
extern "C" void kernel_launch(void* const* d_in, const int* in_sizes, int n_in,
                              void* d_out, int out_size, void* d_ws, size_t ws_size,
                              hipStream_t stream) {
  (void)in_sizes; (void)n_in; (void)out_size; (void)ws_size;
  const float* pred = (const float*)d_in[0];
  float* out = (float*)d_out;

  // Workspace carve-up (~30.1 MB total): ctrl | se_x | se_y | seed | uncl | prop | inst
  char* ws = (char*)d_ws;
  size_t ofs = (sizeof(Ctrl) + 255) & ~(size_t)255;
  Ctrl* c = (Ctrl*)ws;
  float*   sex  = (float*)(ws + ofs);            ofs += (size_t)NPIX * 4;
  float*   sey  = (float*)(ws + ofs);            ofs += (size_t)NPIX * 4;
  float*   sd   = (float*)(ws + ofs);            ofs += (size_t)NPIX * 4;
  uint8_t* uncl = (uint8_t*)(ws + ofs);          ofs += (size_t)NPIX;
  uint8_t* prop = (uint8_t*)(ws + ofs);          ofs += (size_t)NPIX;
  uint8_t* inst = (uint8_t*)(ws + ofs);          ofs += (size_t)NPIX;

  hipLaunchKernelGGL(k_init, dim3(1), dim3(256), 0, stream, c);
  hipLaunchKernelGGL(k_pre, dim3(NPIX / 256), dim3(256), 0, stream,
                     pred, c, sex, sey, sd, uncl, inst);
  hipLaunchKernelGGL(k_cluster, dim3(NB), dim3(NT), 0, stream,
                     pred, out, c, sex, sey, sd, uncl, prop, inst);
}